// BACKFLOW_9775345566267
// MI455X (gfx1250) — compile-verified
//
#include <hip/hip_runtime.h>
#include <hip/hip_bf16.h>

// Problem constants from the reference
#define BB 8192   // batch
#define OO 256    // orbitals
#define EE 64     // electrons (matrix dim)
#define HH 4      // hidden

typedef __attribute__((ext_vector_type(2))) float v2f;
typedef __attribute__((ext_vector_type(8))) float v8f;

#define LDA 65    // padded LDS row stride (conflict-free columns)

// ---------------------------------------------------------------------------
// Kernel 1: per-sample occupied-index extraction + tiny MLP (one wave/sample)
// ---------------------------------------------------------------------------
__global__ __launch_bounds__(256) void BACKFLOW_prep_kernel(
    const float* __restrict__ x,  const float* __restrict__ W1,
    const float* __restrict__ b1, const float* __restrict__ W2,
    const float* __restrict__ b2, int* __restrict__ sel_g,
    float* __restrict__ hvec_g)
{
  const int wave = threadIdx.x >> 5;
  const int lane = threadIdx.x & 31;
  const int b = blockIdx.x * 8 + wave;
  const float* xr = x + (size_t)b * OO;

  float s0 = 0.f, s1 = 0.f, s2 = 0.f, s3 = 0.f;
  int base = 0;
  // 256 orbitals = 8 chunks of 32 lanes; ballot gives ascending ranks like
  // jnp.nonzero(size=E)
  for (int c = 0; c < 8; ++c) {
    const int idx = c * 32 + lane;
    const float v = xr[idx];
    const bool occ = v > 0.5f;
    const unsigned long long m = __ballot(occ);
    const int rank = (int)__popcll(m & ((1ull << lane) - 1ull));
    if (occ) {
      sel_g[(size_t)b * EE + base + rank] = idx;
      s0 += W1[idx * HH + 0];
      s1 += W1[idx * HH + 1];
      s2 += W1[idx * HH + 2];
      s3 += W1[idx * HH + 3];
    }
    base += (int)__popcll(m);
  }
  // wave32 reduce
  for (int off = 16; off > 0; off >>= 1) {
    s0 += __shfl_xor(s0, off);
    s1 += __shfl_xor(s1, off);
    s2 += __shfl_xor(s2, off);
    s3 += __shfl_xor(s3, off);
  }
  const float t0 = fmaxf(s0 + b1[0], 0.f);
  const float t1 = fmaxf(s1 + b1[1], 0.f);
  const float t2 = fmaxf(s2 + b1[2], 0.f);
  const float t3 = fmaxf(s3 + b1[3], 0.f);
  float h[HH];
#pragma unroll
  for (int j = 0; j < HH; ++j) {
    h[j] = fmaxf(t0 * W2[0 * HH + j] + t1 * W2[1 * HH + j] +
                 t2 * W2[2 * HH + j] + t3 * W2[3 * HH + j] + b2[j], 0.f);
  }
  if (lane < HH) hvec_g[(size_t)b * HH + lane] = h[lane];
}

// ---------------------------------------------------------------------------
// Kernel 2: build gathered 64x64 matrix in LDS, blocked LU (NB=16) with
// partial pivoting; trailing updates A22 -= L21*U12 via V_WMMA_F32_16X16X4_F32.
// One wave per sample, 2 samples per 64-thread block. All control flow is
// wave-uniform => EXEC is all ones at every WMMA.
// ---------------------------------------------------------------------------
__global__ __launch_bounds__(64) void BACKFLOW_ludet_kernel(
    const int* __restrict__ sel_g, const float* __restrict__ hvec_g,
    const float* __restrict__ Wg,  const float* __restrict__ bg,
    float* __restrict__ out)
{
  __shared__ float Asm[2][EE * LDA];
  const int wave = threadIdx.x >> 5;
  const int lane = threadIdx.x & 31;
  const int b = blockIdx.x * 2 + wave;
  float* A = Asm[wave];

  const float h0 = hvec_g[(size_t)b * HH + 0];
  const float h1 = hvec_g[(size_t)b * HH + 1];
  const float h2 = hvec_g[(size_t)b * HH + 2];
  const float h3 = hvec_g[(size_t)b * HH + 3];

  // Build rows[i][j] = bg[sel_i][j] + sum_h h[h]*Wg[h][sel_i][j]
  // Wg layout [H][O][E], strides (O*E, E, 1) = (16384, 64, 1); Wg is 256 KB
  // -> L2-resident across the whole grid.
  for (int i = 0; i < EE; ++i) {
    const int o = sel_g[(size_t)b * EE + i];
    const float* w = Wg + (size_t)o * EE;
#pragma unroll
    for (int jj = 0; jj < 2; ++jj) {
      const int j = lane + jj * 32;
      const float v = bg[o * EE + j] + h0 * w[j] + h1 * w[16384 + j] +
                      h2 * w[32768 + j] + h3 * w[49152 + j];
      A[i * LDA + j] = v;
    }
  }
  __syncthreads();

  float sign = 1.0f;  // uniform across lanes
  const int half = lane >> 4;
  const int l16 = lane & 15;

  for (int k = 0; k < 4; ++k) {
    const int kc = k * 16;
    // ---- panel factorization (cols kc..kc+15) with partial pivoting ----
    for (int j = kc; j < kc + 16; ++j) {
      // pivot search over rows j..63 of column j
      float bestv = -1.0f;
      int bestr = j;
#pragma unroll
      for (int rr = 0; rr < 2; ++rr) {
        const int r = lane + rr * 32;
        if (r >= j) {
          const float av = fabsf(A[r * LDA + j]);
          if (av > bestv) { bestv = av; bestr = r; }
        }
      }
      for (int off = 16; off > 0; off >>= 1) {
        const float ov = __shfl_xor(bestv, off);
        const int orr = __shfl_xor(bestr, off);
        if (ov > bestv || (ov == bestv && orr < bestr)) { bestv = ov; bestr = orr; }
      }
      const int p = bestr;            // uniform
      if (p != j) sign = -sign;       // uniform update
      // unconditional full-row swap (p==j is a harmless self-swap)
#pragma unroll
      for (int jj = 0; jj < 2; ++jj) {
        const int c = lane + jj * 32;
        const float tp = A[p * LDA + c];
        const float tj = A[j * LDA + c];
        A[p * LDA + c] = tj;
        A[j * LDA + c] = tp;
      }
      __syncthreads();
      const float rp = 1.0f / A[j * LDA + j];
      // scale subcolumn, rank-1 update restricted to the panel
#pragma unroll
      for (int rr = 0; rr < 2; ++rr) {
        const int r = lane + rr * 32;
        if (r > j) {
          const float l = A[r * LDA + j] * rp;
          A[r * LDA + j] = l;
          for (int c = j + 1; c < kc + 16; ++c)
            A[r * LDA + c] = fmaf(-l, A[j * LDA + c], A[r * LDA + c]);
        }
      }
      __syncthreads();
    }
    if (k == 3) break;

    // ---- triangular solve: U12 = L11^{-1} * A12 (rows kc..kc+15) ----
    for (int jj = 1; jj < 16; ++jj) {
      const int row = kc + jj;
      for (int c = kc + 16 + lane; c < EE; c += 32) {
        float acc = A[row * LDA + c];
        for (int t = 0; t < jj; ++t)
          acc = fmaf(-A[row * LDA + kc + t], A[(kc + t) * LDA + c], acc);
        A[row * LDA + c] = acc;
      }
      __syncthreads();
    }

    // ---- trailing update A22 -= L21 * U12 via WMMA f32 16x16x4 ----
    const int ntiles = 3 - k;  // uniform
    for (int ti = 0; ti < ntiles; ++ti) {
      const int rbase = kc + 16 + ti * 16;
      for (int tj = 0; tj < ntiles; ++tj) {
        const int cbase = kc + 16 + tj * 16;
        // C/D layout: VGPR v holds M = v (lanes 0-15) / M = v+8 (lanes 16-31),
        // N = lane&15
        v8f acc;
#pragma unroll
        for (int v = 0; v < 8; ++v)
          acc[v] = A[(rbase + v + 8 * half) * LDA + cbase + l16];
#pragma unroll
        for (int kk = 0; kk < 4; ++kk) {
          const int k0 = kc + kk * 4 + 2 * half;
          v2f af, bf;
          // A frag (16x4): lane half selects K pair {0,1}/{2,3}; pre-negate
          // since f32 WMMA has no A/B NEG modifier (only C-mod per ISA).
          af.x = -A[(rbase + l16) * LDA + k0];
          af.y = -A[(rbase + l16) * LDA + k0 + 1];
          // B frag (4x16): row K striped across lane halves, col = lane&15
          bf.x = A[k0 * LDA + cbase + l16];
          bf.y = A[(k0 + 1) * LDA + cbase + l16];
          acc = __builtin_amdgcn_wmma_f32_16x16x4_f32(
              false, af, false, bf, (short)0, acc, false, false);
        }
#pragma unroll
        for (int v = 0; v < 8; ++v)
          A[(rbase + v + 8 * half) * LDA + cbase + l16] = acc[v];
      }
    }
    __syncthreads();
  }

  // det = sign * prod(diag(U))
  const float d0 = A[lane * LDA + lane];
  const float d1 = A[(lane + 32) * LDA + (lane + 32)];
  float prod = d0 * d1;
  for (int off = 16; off > 0; off >>= 1)
    prod *= __shfl_xor(prod, off);
  if (lane == 0) out[b] = sign * prod;
}

// ---------------------------------------------------------------------------
extern "C" void kernel_launch(void* const* d_in, const int* in_sizes, int n_in,
                              void* d_out, int out_size, void* d_ws, size_t ws_size,
                              hipStream_t stream) {
  const float* x  = (const float*)d_in[0];
  const float* W1 = (const float*)d_in[1];
  const float* b1 = (const float*)d_in[2];
  const float* W2 = (const float*)d_in[3];
  const float* b2 = (const float*)d_in[4];
  const float* Wg = (const float*)d_in[5];
  const float* bg = (const float*)d_in[6];
  float* out = (float*)d_out;

  int*   sel_g  = (int*)d_ws;                                  // B*64 ints
  float* hvec_g = (float*)((char*)d_ws + (size_t)BB * EE * 4); // B*4 floats

  BACKFLOW_prep_kernel<<<dim3(BB / 8), dim3(256), 0, stream>>>(
      x, W1, b1, W2, b2, sel_g, hvec_g);
  BACKFLOW_ludet_kernel<<<dim3(BB / 2), dim3(64), 0, stream>>>(
      sel_g, hvec_g, Wg, bg, out);
}